// AttentiveErasing_7069516169624
// MI455X (gfx1250) — compile-verified
//
#include <hip/hip_runtime.h>
#include <cstdint>
#include <cfloat>

// ---------------------------------------------------------------------------
// AttentiveErasing for MI455X (gfx1250).
// One workgroup per sample; whole 224x224 fp32 image staged in LDS (200704 B
// out of the WGP's 320 KB) via the async global->LDS path (ASYNCcnt), then:
//   phase 1: max/argmax/min reduction   (wave32 shfl + LDS combine)
//   phase 2: box-bounds reduction of (x > thr)
//   phase 3: elementwise mask/noise/out, NT-hinted b128 stores
// HBM traffic: 100 MB read + 205 MB write  ->  ~13 us at 23.3 TB/s.
// ---------------------------------------------------------------------------

#define BD     256                 // 8 wave32 per workgroup
#define HW     224
#define NPIX   (HW * HW)           // 50176
#define NV4    (NPIX / 4)          // 12544 float4 per image
#define ITERS  (NV4 / BD)          // 49 float4 per thread
#define NBATCH 512
#define ROWV4  (HW / 4)            // 56 float4 per row

typedef __attribute__((ext_vector_type(4))) float v4f;   // true vector type

// --- deterministic hash RNG (stand-in for jax threefry; fixed seeds) --------
__device__ __forceinline__ uint32_t pcg(uint32_t v) {
  v = v * 747796405u + 2891336453u;
  v = ((v >> ((v >> 28u) + 4u)) ^ v) * 277803737u;
  return (v >> 22u) ^ v;
}
__device__ __forceinline__ float u01(uint32_t h) {
  return (float)(h >> 8) * (1.0f / 16777216.0f);   // [0,1)
}

// --- gfx1250 async global -> LDS (tracked by ASYNCcnt) ----------------------
__device__ __forceinline__ void async_ld_b128(uint32_t lds_byte_addr, const void* gaddr) {
  asm volatile("global_load_async_to_lds_b128 %0, %1, off"
               :: "v"(lds_byte_addr), "v"(gaddr) : "memory");
}
__device__ __forceinline__ void wait_async0() {
#if __has_builtin(__builtin_amdgcn_s_wait_asynccnt)
  __builtin_amdgcn_s_wait_asynccnt(0);
#else
  asm volatile("s_wait_asynccnt 0x0" ::: "memory");
#endif
}

__global__ __launch_bounds__(BD) void attentive_erase_kernel(
    const float* __restrict__ x, float* __restrict__ out, float* __restrict__ maskO) {
  __shared__ float s_img[NPIX];                 // 200704 bytes
  __shared__ float s_mx[8];
  __shared__ int   s_mi[8];
  __shared__ float s_mn[8];
  __shared__ int   s_b0[8], s_b1[8], s_b2[8], s_b3[8];
  __shared__ float s_thr;
  __shared__ int   s_amax;
  __shared__ int   s_box[5];                    // hs, he, ws, we, do

  const int b    = blockIdx.x;
  const int tid  = threadIdx.x;
  const int lane = tid & 31;
  const int wv   = tid >> 5;

  // ---- stage image into LDS (async path, no VGPR round trip) ----
  const v4f* gx = (const v4f*)(x + (size_t)b * NPIX);
  for (int k = 0; k < ITERS; ++k) {
    const int v = k * BD + tid;
    async_ld_b128((uint32_t)(uintptr_t)(&s_img[v * 4]), (const void*)(gx + v));
  }
  wait_async0();
  __syncthreads();

  const v4f* l4 = (const v4f*)s_img;

  // ---- phase 1: max (first-occurrence argmax) + min ----
  float mx = -FLT_MAX, mn = FLT_MAX;
  int   mi = 0;
  for (int k = 0; k < ITERS; ++k) {
    const int v = k * BD + tid;
    const v4f f = l4[v];
    const int e = v * 4;
    if (f.x > mx || (f.x == mx && e     < mi)) { mx = f.x; mi = e;     }
    if (f.y > mx || (f.y == mx && e + 1 < mi)) { mx = f.y; mi = e + 1; }
    if (f.z > mx || (f.z == mx && e + 2 < mi)) { mx = f.z; mi = e + 2; }
    if (f.w > mx || (f.w == mx && e + 3 < mi)) { mx = f.w; mi = e + 3; }
    mn = fminf(mn, fminf(fminf(f.x, f.y), fminf(f.z, f.w)));
  }
  for (int off = 16; off > 0; off >>= 1) {      // wave32 butterfly
    const float ov = __shfl_xor(mx, off, 32);
    const int   oi = __shfl_xor(mi, off, 32);
    const float on = __shfl_xor(mn, off, 32);
    if (ov > mx || (ov == mx && oi < mi)) { mx = ov; mi = oi; }
    mn = fminf(mn, on);
  }
  if (lane == 0) { s_mx[wv] = mx; s_mi[wv] = mi; s_mn[wv] = mn; }
  __syncthreads();
  if (tid == 0) {
    for (int i = 1; i < 8; ++i) {
      if (s_mx[i] > mx || (s_mx[i] == mx && s_mi[i] < mi)) { mx = s_mx[i]; mi = s_mi[i]; }
      mn = fminf(mn, s_mn[i]);
    }
    const float factor = 0.5f * u01(pcg(42u ^ 0xA511E9B3u));   // fixed root key
    s_thr  = mx - (mx - mn) * factor;
    s_amax = mi;
  }
  __syncthreads();
  const float thr = s_thr;

  // ---- phase 2: bounds of prop = (x > thr) ----
  int miny = HW, maxy = -1, minx = HW, maxx = -1;
  for (int k = 0; k < ITERS; ++k) {
    const int v  = k * BD + tid;
    const v4f f = l4[v];
    const int yy = v / ROWV4;
    const int xb = (v % ROWV4) * 4;
    bool rowAny = false;
    if (f.x > thr) { minx = min(minx, xb);     maxx = max(maxx, xb);     rowAny = true; }
    if (f.y > thr) { minx = min(minx, xb + 1); maxx = max(maxx, xb + 1); rowAny = true; }
    if (f.z > thr) { minx = min(minx, xb + 2); maxx = max(maxx, xb + 2); rowAny = true; }
    if (f.w > thr) { minx = min(minx, xb + 3); maxx = max(maxx, xb + 3); rowAny = true; }
    if (rowAny) { miny = min(miny, yy); maxy = max(maxy, yy); }
  }
  for (int off = 16; off > 0; off >>= 1) {
    miny = min(miny, __shfl_xor(miny, off, 32));
    maxy = max(maxy, __shfl_xor(maxy, off, 32));
    minx = min(minx, __shfl_xor(minx, off, 32));
    maxx = max(maxx, __shfl_xor(maxx, off, 32));
  }
  if (lane == 0) { s_b0[wv] = miny; s_b1[wv] = maxy; s_b2[wv] = minx; s_b3[wv] = maxx; }
  __syncthreads();
  if (tid == 0) {
    for (int i = 1; i < 8; ++i) {
      miny = min(miny, s_b0[i]); maxy = max(maxy, s_b1[i]);
      minx = min(minx, s_b2[i]); maxx = max(maxx, s_b3[i]);
    }
    const int  cy = s_amax / HW, cx = s_amax % HW;
    const int  mh2 = (maxy - miny) / 2;
    const int  mw2 = (maxx - minx) / 2;
    const bool any_prop = (maxy >= 0);
    const bool valid = any_prop && (mh2 > 4) && (mw2 > 4);
    const uint32_t hb = pcg(0x85EBCA6Bu + (uint32_t)b * 2654435761u);
    const bool applyf = u01(pcg(hb ^ 0x1u)) < 0.5f;
    const int  hh = 4 + (int)(u01(pcg(hb ^ 0x2u)) * (float)max(mh2 - 4, 1));
    const int  ww = 4 + (int)(u01(pcg(hb ^ 0x3u)) * (float)max(mw2 - 4, 1));
    s_box[0] = max(cy - hh, 0);
    s_box[1] = min(cy + hh, HW);
    s_box[2] = max(cx - ww, 0);
    s_box[3] = min(cx + ww, HW);
    s_box[4] = (applyf && valid) ? 1 : 0;
  }
  __syncthreads();

  // ---- phase 3: elementwise out + mask, NT stores ----
  const int hs = s_box[0], he = s_box[1], wsx = s_box[2], we = s_box[3];
  const int dof = s_box[4];
  v4f* o4 = (v4f*)(out   + (size_t)b * NPIX);
  v4f* m4 = (v4f*)(maskO + (size_t)b * NPIX);
  for (int k = 0; k < ITERS; ++k) {
    const int v = k * BD + tid;
    const v4f f = l4[v];
    const int yy = v / ROWV4;
    const int xb = (v % ROWV4) * 4;
    const bool yok = (yy > hs) && (yy < he) && (yy > 0);
    const uint32_t ebase = (uint32_t)(b * NPIX + v * 4);
    float vals[4] = {f.x, f.y, f.z, f.w};
    float ov[4], mv[4];
#pragma unroll
    for (int c = 0; c < 4; ++c) {
      const float val  = vals[c];
      const int   xx   = xb + c;
      const bool  prop = val > thr;
      const bool  erase = prop && yok && (xx > wsx) && (xx < we) && (xx > 0);
      const float maskv = (dof && erase) ? 0.0f : 1.0f;
      const float inp   = 0.6f * val + 0.2f;
      const float keep  = fminf(fmaxf(1.0f - inp, 0.0f), 1.0f);
      const float u     = u01(pcg(pcg(ebase + (uint32_t)c) ^ 0x9E3779B9u));
      const float pm    = (u < keep) ? 1.0f : 0.0f;
      const float noise = maskv + (1.0f - maskv) * pm;
      ov[c] = inp * noise;
      mv[c] = maskv;
    }
    const v4f ovv = {ov[0], ov[1], ov[2], ov[3]};
    const v4f mvv = {mv[0], mv[1], mv[2], mv[3]};
    __builtin_nontemporal_store(ovv, o4 + v);
    __builtin_nontemporal_store(mvv, m4 + v);
  }
}

extern "C" void kernel_launch(void* const* d_in, const int* in_sizes, int n_in,
                              void* d_out, int out_size, void* d_ws, size_t ws_size,
                              hipStream_t stream) {
  const float* x   = (const float*)d_in[0];
  float*       out = (float*)d_out;
  float*       msk = out + (size_t)NBATCH * NPIX;   // tuple (out, mask) concatenated
  hipLaunchKernelGGL(attentive_erase_kernel, dim3(NBATCH), dim3(BD), 0, stream,
                     x, out, msk);
}